// BaseModel_55705725829328
// MI455X (gfx1250) — compile-verified
//
#include <hip/hip_runtime.h>

typedef float v2f __attribute__((ext_vector_type(2)));
typedef float v8f __attribute__((ext_vector_type(8)));

#define DVARS 100
#define HDIM  16
#define PDIM  2
#define BSZ   4096
#define BT    16      // batch tile per workgroup
#define NWAVE 8
#define NTHR  256

// workspace layout (in floats)
#define W0B_OFF   0
#define W0B_ELEMS (DVARS * 25 * 64)   // 160000 : B-layout W0, K=100 -> 25 chunks of 4
#define W1B_OFF   (W0B_OFF + W0B_ELEMS)
#define W1B_ELEMS (DVARS * 4 * 64)    // 25600  : B-layout W1, K=16 -> 4 chunks
#define W2B_OFF   (W1B_OFF + W1B_ELEMS)
#define W2B_ELEMS (DVARS * 4 * 64)    // 25600  : B-layout W2 (N padded 2->16 with zeros)

// ---------------------------------------------------------------------------
// Rewrite W0/W1/W2 into WMMA B-operand layout: element i = ((t*C + c)*32 + lane)*2 + v
// holds B[k = 4c + 2*(lane>>4) + v, n = lane&15] for tile t.
// ---------------------------------------------------------------------------
__global__ void prep_kernel(const float* __restrict__ W0,
                            const float* __restrict__ W1,
                            const float* __restrict__ W2,
                            float* __restrict__ ws) {
  float* w0b = ws + W0B_OFF;
  float* w1b = ws + W1B_OFF;
  float* w2b = ws + W2B_OFF;
  const int stride = gridDim.x * blockDim.x;
  const int idx0   = blockIdx.x * blockDim.x + threadIdx.x;

  for (int i = idx0; i < W0B_ELEMS; i += stride) {
    int v    = i & 1;
    int lane = (i >> 1) & 31;
    int tc   = i >> 6;
    int t    = tc / 25;
    int c    = tc - t * 25;
    int n    = lane & 15;
    int k    = 4 * c + 2 * (lane >> 4) + v;          // 0..99
    w0b[i] = W0[(t * HDIM + n) * DVARS + k];         // W0[t, i=n, j=k]
  }
  for (int i = idx0; i < W1B_ELEMS; i += stride) {
    int v    = i & 1;
    int lane = (i >> 1) & 31;
    int tc   = i >> 6;
    int t    = tc >> 2;
    int c    = tc & 3;
    int n    = lane & 15;
    int k    = 4 * c + 2 * (lane >> 4) + v;          // 0..15
    w1b[i] = W1[(t * HDIM + n) * HDIM + k];          // W1[t, i=n, j=k]
  }
  for (int i = idx0; i < W2B_ELEMS; i += stride) {
    int v    = i & 1;
    int lane = (i >> 1) & 31;
    int tc   = i >> 6;
    int t    = tc >> 2;
    int c    = tc & 3;
    int n    = lane & 15;
    int k    = 4 * c + 2 * (lane >> 4) + v;
    w2b[i] = (n < PDIM) ? W2[(t * PDIM + n) * HDIM + k] : 0.0f;   // pad N to 16
  }
}

// ---------------------------------------------------------------------------
// Main fused kernel: one workgroup per 16-batch tile; 8 waves sweep t = 0..99.
// ---------------------------------------------------------------------------
__launch_bounds__(NTHR)
__global__ void dagnn_kernel(const float* __restrict__ x,
                             const float* __restrict__ la,
                             const float* __restrict__ noise,
                             const float* __restrict__ b0,
                             const float* __restrict__ b1,
                             const float* __restrict__ b2,
                             const float* __restrict__ ws,
                             float* __restrict__ out) {
  __shared__ float    xs[BT * 101];           // x tile, padded rows (odd stride)
  __shared__ unsigned maskw[BT * 401];        // t-packed mask words [bi][j*4+w], padded
  __shared__ float    hs[NWAVE * 16 * 17];    // per-wave 16x16 h tile, padded

  const float* __restrict__ w0b = ws + W0B_OFF;
  const float* __restrict__ w1b = ws + W1B_OFF;
  const float* __restrict__ w2b = ws + W2B_OFF;

  const int tid  = threadIdx.x;
  const int bOff = blockIdx.x * BT;

  // ---- Phase 1a: stage x tile ----
  for (int u = tid; u < BT * DVARS; u += NTHR) {
    int bi = u / DVARS;
    int j  = u - bi * DVARS;
    xs[bi * 101 + j] = x[(size_t)(bOff + bi) * DVARS + j];
  }

  // ---- Phase 1b: stream noise slab (coalesced f4), pack mask bits over t ----
  for (int u = tid; u < BT * DVARS * 4; u += NTHR) {
    int bi    = u / 400;
    int rem   = u - bi * 400;
    int j     = rem >> 2;
    int w     = rem & 3;
    int tbase = w << 5;
    int nt    = (w == 3) ? 4 : 32;
    const float* np = noise + ((size_t)(bOff + bi) * DVARS + j) * DVARS + tbase;
    const float* lp = la + j * DVARS + tbase;
    unsigned word = 0u;
    for (int k = 0; k < nt; k += 4) {
      float4 nv = *(const float4*)(np + k);
      float4 lv = *(const float4*)(lp + k);
      int t0 = tbase + k;
      if (((nv.x + lv.x) > 0.0f) && (j != t0))     word |= (1u << (k + 0));
      if (((nv.y + lv.y) > 0.0f) && (j != t0 + 1)) word |= (1u << (k + 1));
      if (((nv.z + lv.z) > 0.0f) && (j != t0 + 2)) word |= (1u << (k + 2));
      if (((nv.w + lv.w) > 0.0f) && (j != t0 + 3)) word |= (1u << (k + 3));
    }
    maskw[bi * 401 + rem] = word;
  }
  __syncthreads();

  // ---- Phase 2: WMMA pipeline, one 16x16 tile per (b-tile, t) ----
  const int lane = tid & 31;
  const int wave = tid >> 5;
  const int n16  = lane & 15;   // N (hidden index) / A-matrix M row
  const int hi   = lane >> 4;   // half-wave select
  float* hsw = hs + wave * (16 * 17);

  for (int t = wave; t < DVARS; t += NWAVE) {
    const int      tw = t >> 5;
    const unsigned tb = (unsigned)(t & 31);

    // ----- Layer 0: 16x16x100 GEMM, A = masked x, C init = bias -----
    float bias0 = b0[t * HDIM + n16];
    v8f acc;
    #pragma unroll
    for (int r = 0; r < 8; ++r) acc[r] = bias0;

    const float* bptr0 = w0b + (size_t)t * 25 * 64 + lane * 2;
    #pragma unroll
    for (int c = 0; c < 25; ++c) {
      int j0 = 4 * c + 2 * hi;
      float    x0 = xs[n16 * 101 + j0];
      float    x1 = xs[n16 * 101 + j0 + 1];
      unsigned m0 = maskw[n16 * 401 + j0 * 4 + tw];
      unsigned m1 = maskw[n16 * 401 + (j0 + 1) * 4 + tw];
      v2f A;
      A.x = ((m0 >> tb) & 1u) ? x0 : 0.0f;
      A.y = ((m1 >> tb) & 1u) ? x1 : 0.0f;
      v2f B = *(const v2f*)(bptr0 + c * 64);
      acc = __builtin_amdgcn_wmma_f32_16x16x4_f32(false, A, false, B,
                                                  (short)0, acc, false, false);
    }
    #pragma unroll
    for (int r = 0; r < 8; ++r) acc[r] = fmaxf(acc[r], 0.01f * acc[r]);  // leaky_relu

    // C-layout -> A-layout via per-wave LDS round trip (per-wave LDS is in-order)
    asm volatile("" ::: "memory");
    #pragma unroll
    for (int r = 0; r < 8; ++r) hsw[(r + 8 * hi) * 17 + n16] = acc[r];
    asm volatile("s_wait_dscnt 0" ::: "memory");

    // ----- Layer 1: 16x16x16 GEMM -----
    float bias1 = b1[t * HDIM + n16];
    v8f acc1;
    #pragma unroll
    for (int r = 0; r < 8; ++r) acc1[r] = bias1;
    const float* bptr1 = w1b + (size_t)t * 4 * 64 + lane * 2;
    #pragma unroll
    for (int c = 0; c < 4; ++c) {
      int j0 = 4 * c + 2 * hi;
      v2f A;
      A.x = hsw[n16 * 17 + j0];
      A.y = hsw[n16 * 17 + j0 + 1];
      v2f B = *(const v2f*)(bptr1 + c * 64);
      acc1 = __builtin_amdgcn_wmma_f32_16x16x4_f32(false, A, false, B,
                                                   (short)0, acc1, false, false);
    }
    #pragma unroll
    for (int r = 0; r < 8; ++r) acc1[r] = fmaxf(acc1[r], 0.01f * acc1[r]);

    asm volatile("" ::: "memory");
    #pragma unroll
    for (int r = 0; r < 8; ++r) hsw[(r + 8 * hi) * 17 + n16] = acc1[r];
    asm volatile("s_wait_dscnt 0" ::: "memory");

    // ----- Layer 2: 16x16x16 GEMM (only N=0,1 valid) -----
    float bias2 = (n16 < PDIM) ? b2[t * PDIM + n16] : 0.0f;
    v8f acc2;
    #pragma unroll
    for (int r = 0; r < 8; ++r) acc2[r] = bias2;
    const float* bptr2 = w2b + (size_t)t * 4 * 64 + lane * 2;
    #pragma unroll
    for (int c = 0; c < 4; ++c) {
      int j0 = 4 * c + 2 * hi;
      v2f A;
      A.x = hsw[n16 * 17 + j0];
      A.y = hsw[n16 * 17 + j0 + 1];
      v2f B = *(const v2f*)(bptr2 + c * 64);
      acc2 = __builtin_amdgcn_wmma_f32_16x16x4_f32(false, A, false, B,
                                                   (short)0, acc2, false, false);
    }
    asm volatile("" ::: "memory");

    if (n16 < PDIM) {
      #pragma unroll
      for (int r = 0; r < 8; ++r) {
        out[((size_t)(bOff + r + 8 * hi) * DVARS + t) * PDIM + n16] = acc2[r];
      }
    }
  }
}

extern "C" void kernel_launch(void* const* d_in, const int* in_sizes, int n_in,
                              void* d_out, int out_size, void* d_ws, size_t ws_size,
                              hipStream_t stream) {
  const float* x     = (const float*)d_in[0];
  const float* la    = (const float*)d_in[1];
  const float* noise = (const float*)d_in[2];
  const float* W0    = (const float*)d_in[3];
  const float* b0    = (const float*)d_in[4];
  const float* W1    = (const float*)d_in[5];
  const float* b1    = (const float*)d_in[6];
  const float* W2    = (const float*)d_in[7];
  const float* b2    = (const float*)d_in[8];
  float* out = (float*)d_out;
  float* ws  = (float*)d_ws;

  prep_kernel<<<240, 256, 0, stream>>>(W0, W1, W2, ws);
  dagnn_kernel<<<BSZ / BT, NTHR, 0, stream>>>(x, la, noise, b0, b1, b2, ws, out);
}